// kernel_loss_90915867721752
// MI455X (gfx1250) — compile-verified
//
#include <hip/hip_runtime.h>

typedef __attribute__((ext_vector_type(2))) float v2f;
typedef __attribute__((ext_vector_type(8))) float v8f;

#define HH   256
#define WW   256
#define HO   248
#define WO   248
#define NBAT 4
#define TILE 16
#define HALO 8
#define LDSW (TILE + HALO)   // 24

// One block = one 16x16 output tile of one batch image.
// LDS: 24x24 halo tile; channels (x0,x1,x2,y0) packed in float4 so each
// neighbor costs one ds_load_b128 + one ds_load_b32 (y1) instead of 5x b32.
__global__ __launch_bounds__(256) void potts_kernel(
    const float* __restrict__ x, const float* __restrict__ y,
    float* __restrict__ partials)
{
    __shared__ float4 t4[LDSW][LDSW];   // x0, x1, x2, y0
    __shared__ float  t1[LDSW][LDSW];   // y1
    __shared__ float  wsum[8];

    const int b   = blockIdx.z;
    const int iy0 = blockIdx.y * TILE;
    const int ix0 = blockIdx.x * TILE;
    const int tx  = threadIdx.x, ty = threadIdx.y;
    const int tid = ty * TILE + tx;

    const float* xb = x + (size_t)b * 3 * HH * WW;
    const float* yb = y + (size_t)b * 2 * HH * WW;
    const int HW = HH * WW;

    // Cooperative load of the 24x24 halo tile (zero-fill out of range; those
    // entries are only touched by invalid output pixels whose acc is masked).
    for (int idx = tid; idx < LDSW * LDSW; idx += 256) {
        const int lr = idx / LDSW, lc = idx % LDSW;
        const int r = iy0 + lr, c = ix0 + lc;
        float4 v = make_float4(0.0f, 0.0f, 0.0f, 0.0f);
        float  vy1 = 0.0f;
        if (r < HH && c < WW) {
            const size_t off = (size_t)r * WW + c;
            v.x = xb[0 * HW + off];
            v.y = xb[1 * HW + off];
            v.z = xb[2 * HW + off];
            v.w = yb[0 * HW + off];
            vy1 = yb[1 * HW + off];
        }
        t4[lr][lc] = v;
        t1[lr][lc] = vy1;
    }
    __syncthreads();

    const int oh = iy0 + ty, ow = ix0 + tx;

    const float4 cc  = t4[ty + 4][tx + 4];   // cx0, cx1, cx2, y0c
    const float  y1c = t1[ty + 4][tx + 4];

    // W = exp(-color/sigma_I^2) * exp(-(di^2+dj^2)/sigma_X^2)
    //   = exp(-(100*color + (di^2+dj^2)/9))
    float acc = 0.0f;
    const float INV9 = 1.0f / 9.0f;
    for (int i = 0; i < 9; ++i) {
        const float di = (float)(i - 4);
        const float ri = di * di * INV9;
        const float4* __restrict__ rp  = &t4[ty + i][tx];
        const float*  __restrict__ rp1 = &t1[ty + i][tx];
#pragma unroll
        for (int j = 0; j < 9; ++j) {
            const float dj = (float)(j - 4);
            const float rj = dj * dj * INV9;
            const float4 nb  = rp[j];       // ds_load_b128
            const float  ny1 = rp1[j];      // ds_load_b32
            const float d0 = cc.x - nb.x;
            const float d1 = cc.y - nb.y;
            const float d2 = cc.z - nb.z;
            const float color = d0 * d0 + d1 * d1 + d2 * d2;
            const float w = __expf(-(100.0f * color + ri + rj));
            acc += w * (cc.w * ny1 + y1c * nb.w);
        }
    }
    // Mask pixels outside the 248x248 valid output region.
    acc = (oh < HO && ow < WO) ? acc : 0.0f;

    // --- Wave-level reduction via WMMA ---------------------------------
    // A (16x4 f32, 2 VGPRs/lane): A[m][0] = acc(lane m), A[m][2] = acc(lane m+16),
    // other K entries zero. B = all-ones => D[m][n] = acc(m) + acc(m+16).
    // Lane 0 column holds M=0..7 in d[0..7]; lane 16 column holds M=8..15.
    v2f a;  a[0] = acc;  a[1] = 0.0f;
    v2f bone; bone[0] = 1.0f; bone[1] = 1.0f;
    v8f cz = {0.0f, 0.0f, 0.0f, 0.0f, 0.0f, 0.0f, 0.0f, 0.0f};
    v8f d = __builtin_amdgcn_wmma_f32_16x16x4_f32(
        false, a, false, bone, (short)0, cz, false, false);
    float s = d[0] + d[1] + d[2] + d[3] + d[4] + d[5] + d[6] + d[7];
    const float wavetotal = __shfl(s, 0, 32) + __shfl(s, 16, 32);

    const int lane = tid & 31, wave = tid >> 5;
    if (lane == 0) wsum[wave] = wavetotal;
    __syncthreads();

    if (tid == 0) {
        float bt = 0.0f;
        for (int wv = 0; wv < 8; ++wv) bt += wsum[wv];
        partials[((size_t)blockIdx.z * gridDim.y + blockIdx.y) * gridDim.x +
                 blockIdx.x] = bt;
    }
}

// Deterministic fixed-order final reduction (double accumulator), then mean.
__global__ void finalize_kernel(const float* __restrict__ partials,
                                float* __restrict__ out, int n)
{
    if (threadIdx.x == 0 && blockIdx.x == 0) {
        double s = 0.0;
        for (int i = 0; i < n; ++i) s += (double)partials[i];
        const double inv = 1.0 / ((double)NBAT * 81.0 * (double)HO * (double)WO);
        out[0] = (float)(s * inv);
    }
}

extern "C" void kernel_launch(void* const* d_in, const int* in_sizes, int n_in,
                              void* d_out, int out_size, void* d_ws, size_t ws_size,
                              hipStream_t stream) {
    const float* x = (const float*)d_in[0];   // [4,3,256,256] f32
    const float* y = (const float*)d_in[1];   // [4,2,256,256] f32
    float* out = (float*)d_out;               // scalar f32
    float* partials = (float*)d_ws;           // 1024 floats of scratch

    dim3 grid((WO + TILE - 1) / TILE, (HO + TILE - 1) / TILE, NBAT); // 16x16x4
    dim3 block(TILE, TILE);
    potts_kernel<<<grid, block, 0, stream>>>(x, y, partials);

    const int nblk = grid.x * grid.y * grid.z; // 1024
    finalize_kernel<<<1, 32, 0, stream>>>(partials, out, nblk);
}